// shared_encoder_26843545600017
// MI455X (gfx1250) — compile-verified
//
#include <hip/hip_runtime.h>
#include <hip/hip_bf16.h>

typedef float v2f __attribute__((ext_vector_type(2)));
typedef float v8f __attribute__((ext_vector_type(8)));

#define IN_FEATS 256
#define OUT_DIM 64

// ---------------------------------------------------------------------------
// 1) init: deg = 1.0 (self loop), agg = 0
// ---------------------------------------------------------------------------
__global__ void gcn_init_kernel(float* __restrict__ deg, float* __restrict__ agg,
                                int n_nodes) {
  long long i = (long long)blockIdx.x * blockDim.x + threadIdx.x;
  if (i < n_nodes) deg[i] = 1.0f;
  if (i < (long long)n_nodes * OUT_DIM) agg[i] = 0.0f;
}

// ---------------------------------------------------------------------------
// 2) h = x @ W  via V_WMMA_F32_16X16X4_F32 (fp32, exact)
//    One wave computes a 16x64 tile of h (4 column tiles of 16).
//    A 16x4 f32 frag: lane%16 = M row, K = 2*(lane/16) + vgpr
//    B 4x16 f32 frag: lane%16 = N col, K = 2*(lane/16) + vgpr
//    D 16x16 f32:     lane%16 = N col, M = vgpr + 8*(lane/16)
// ---------------------------------------------------------------------------
__global__ void gcn_gemm_wmma_kernel(const float* __restrict__ x,
                                     const float* __restrict__ W,
                                     float* __restrict__ h, int n_rows) {
  const int wave = (blockIdx.x * blockDim.x + threadIdx.x) >> 5;
  const int lane = threadIdx.x & 31;
  const int row_tile = wave;
  if (row_tile * 16 >= n_rows) return;  // uniform per wave

  const int g = lane >> 4;   // half-wave group
  const int l = lane & 15;   // row (A) / col (B,D) within tile
  const int row = row_tile * 16 + l;

  v8f acc0 = {}, acc1 = {}, acc2 = {}, acc3 = {};
  const float* xrow = x + (size_t)row * IN_FEATS;

  for (int k = 0; k < IN_FEATS; k += 4) {
    const int ka = k + 2 * g;
    v2f a;
    a.x = xrow[ka];
    a.y = xrow[ka + 1];

    v2f b0, b1, b2, b3;
    const float* w0 = W + (size_t)ka * OUT_DIM + l;
    const float* w1 = w0 + OUT_DIM;
    b0.x = w0[0];  b0.y = w1[0];
    b1.x = w0[16]; b1.y = w1[16];
    b2.x = w0[32]; b2.y = w1[32];
    b3.x = w0[48]; b3.y = w1[48];

    acc0 = __builtin_amdgcn_wmma_f32_16x16x4_f32(false, a, false, b0, (short)0, acc0, false, false);
    acc1 = __builtin_amdgcn_wmma_f32_16x16x4_f32(false, a, false, b1, (short)0, acc1, false, false);
    acc2 = __builtin_amdgcn_wmma_f32_16x16x4_f32(false, a, false, b2, (short)0, acc2, false, false);
    acc3 = __builtin_amdgcn_wmma_f32_16x16x4_f32(false, a, false, b3, (short)0, acc3, false, false);
  }

#pragma unroll
  for (int v = 0; v < 8; ++v) {
    const size_t r = (size_t)(row_tile * 16 + v + 8 * g) * OUT_DIM + l;
    h[r +  0] = acc0[v];
    h[r + 16] = acc1[v];
    h[r + 32] = acc2[v];
    h[r + 48] = acc3[v];
  }
}

// ---------------------------------------------------------------------------
// 3) degree accumulation over edge destinations
// ---------------------------------------------------------------------------
__global__ void gcn_degree_kernel(const int* __restrict__ ei, float* __restrict__ deg,
                                  int n_edges) {
  int e = blockIdx.x * blockDim.x + threadIdx.x;
  if (e < n_edges) atomicAdd(&deg[ei[n_edges + e]], 1.0f);
}

// ---------------------------------------------------------------------------
// 4) dinv = rsqrt(deg)   (deg >= 1 always, self-loops included)
// ---------------------------------------------------------------------------
__global__ void gcn_dinv_kernel(const float* __restrict__ deg, float* __restrict__ dinv,
                                int n_nodes) {
  int i = blockIdx.x * blockDim.x + threadIdx.x;
  if (i < n_nodes) dinv[i] = rsqrtf(deg[i]);
}

// ---------------------------------------------------------------------------
// 5) edge scatter: agg[dst] += h[src] * dinv[src]*dinv[dst]
//    16 threads per edge, 4 floats per thread (float4 gather)
// ---------------------------------------------------------------------------
__global__ void gcn_scatter_kernel(const int* __restrict__ ei,
                                   const float* __restrict__ h,
                                   const float* __restrict__ dinv,
                                   float* __restrict__ agg, int n_edges) {
  long long t = (long long)blockIdx.x * blockDim.x + threadIdx.x;
  long long e = t >> 4;
  if (e >= n_edges) return;
  const int part = (int)(t & 15);
  const int s = ei[e];
  const int d = ei[n_edges + e];
  const float nrm = dinv[s] * dinv[d];
  const float4 hv = ((const float4*)(h + (size_t)s * OUT_DIM))[part];
  float* ap = agg + (size_t)d * OUT_DIM + part * 4;
  atomicAdd(ap + 0, hv.x * nrm);
  atomicAdd(ap + 1, hv.y * nrm);
  atomicAdd(ap + 2, hv.z * nrm);
  atomicAdd(ap + 3, hv.w * nrm);
}

// ---------------------------------------------------------------------------
// 6) out = relu(agg + h*dinv^2 + b)   (self-loop message folded in here)
// ---------------------------------------------------------------------------
__global__ void gcn_finalize_kernel(const float* __restrict__ agg,
                                    const float* __restrict__ h,
                                    const float* __restrict__ dinv,
                                    const float* __restrict__ bias,
                                    float* __restrict__ out, int n_nodes) {
  long long i = (long long)blockIdx.x * blockDim.x + threadIdx.x;
  if (i >= (long long)n_nodes * OUT_DIM) return;
  const int node = (int)(i >> 6);
  const int j = (int)(i & 63);
  const float di = dinv[node];
  const float v = agg[i] + h[i] * di * di + bias[j];
  out[i] = fmaxf(v, 0.0f);
}

extern "C" void kernel_launch(void* const* d_in, const int* in_sizes, int n_in,
                              void* d_out, int out_size, void* d_ws, size_t ws_size,
                              hipStream_t stream) {
  const float* x    = (const float*)d_in[0];
  const int*   ei   = (const int*)d_in[1];   // [2, E] : row0=src, row1=dst
  const float* W    = (const float*)d_in[2];
  const float* bias = (const float*)d_in[3];
  float* out = (float*)d_out;

  const int n_nodes = in_sizes[0] / IN_FEATS;
  const int n_edges = in_sizes[1] / 2;

  // workspace layout (floats)
  float* h    = (float*)d_ws;                         // n_nodes*64
  float* agg  = h + (size_t)n_nodes * OUT_DIM;        // n_nodes*64
  float* deg  = agg + (size_t)n_nodes * OUT_DIM;      // n_nodes
  float* dinv = deg + n_nodes;                        // n_nodes

  const int B = 256;
  const long long nfeat = (long long)n_nodes * OUT_DIM;

  // 1) init
  {
    int grid = (int)((nfeat + B - 1) / B);
    gcn_init_kernel<<<grid, B, 0, stream>>>(deg, agg, n_nodes);
  }
  // 2) GEMM (one wave per 16-row tile)
  {
    int row_tiles = (n_nodes + 15) / 16;
    long long threads = (long long)row_tiles * 32;
    int grid = (int)((threads + B - 1) / B);
    gcn_gemm_wmma_kernel<<<grid, B, 0, stream>>>(x, W, h, n_nodes);
  }
  // 3) degree
  {
    int grid = (n_edges + B - 1) / B;
    gcn_degree_kernel<<<grid, B, 0, stream>>>(ei, deg, n_edges);
  }
  // 4) dinv
  {
    int grid = (n_nodes + B - 1) / B;
    gcn_dinv_kernel<<<grid, B, 0, stream>>>(deg, dinv, n_nodes);
  }
  // 5) scatter
  {
    long long threads = (long long)n_edges * 16;
    int grid = (int)((threads + B - 1) / B);
    gcn_scatter_kernel<<<grid, B, 0, stream>>>(ei, h, dinv, agg, n_edges);
  }
  // 6) finalize
  {
    int grid = (int)((nfeat + B - 1) / B);
    gcn_finalize_kernel<<<grid, B, 0, stream>>>(agg, h, dinv, bias, out, n_nodes);
  }
}